// AttentionAggregator_85186381349610
// MI455X (gfx1250) — compile-verified
//
#include <hip/hip_runtime.h>
#include <hip/hip_bf16.h>

#define N_NODES 50000
#define IN_DIM  256
#define OUT_DIM 128
#define N_EDGES 1600000
#define SLOPE   0.1f

typedef float v2f __attribute__((ext_vector_type(2)));
typedef float v8f __attribute__((ext_vector_type(8)));

// ---------------------------------------------------------------------------
// Zero-init kernel (agg lives directly in d_out; row_sum in workspace).
// ---------------------------------------------------------------------------
__global__ void gat_zero_kernel(float* __restrict__ p, long long n) {
    long long i = (long long)blockIdx.x * blockDim.x + threadIdx.x;
    long long stride = (long long)gridDim.x * blockDim.x;
    for (; i < n; i += stride) p[i] = 0.0f;
}

// ---------------------------------------------------------------------------
// h = features @ W + b  via V_WMMA_F32_16X16X4_F32 (full fp32 precision).
// Block = 128 threads = 4 waves. Block owns one 16-row M tile; wave w owns
// N-tiles {2w, 2w+1} (OUT_DIM = 128 = 8 tiles of 16).
// A 16x4 f32 frag: lanes 0-15 row M=lane, VGPR0=K0/VGPR1=K1; lanes 16-31 K2/K3.
// B 4x16 f32 frag: mirrored (low lanes rows K0/K1, high lanes K2/K3, col=lane&15).
// C/D 16x16 f32: VGPR v -> row v (lanes 0-15) / v+8 (lanes 16-31), col lane&15.
// ---------------------------------------------------------------------------
__global__ __launch_bounds__(128) void gat_gemm_kernel(
    const float* __restrict__ features,   // [N_NODES, IN_DIM]
    const float* __restrict__ W,          // [IN_DIM, OUT_DIM]
    const float* __restrict__ bias,       // [OUT_DIM]
    float* __restrict__ h)                // [N_NODES, OUT_DIM]
{
    const int lane = threadIdx.x & 31;
    const int wave = threadIdx.x >> 5;
    const int half = lane >> 4;           // 0 or 1
    const int col  = lane & 15;

    const int mBase = blockIdx.x * 16;    // 3125 blocks * 16 = 50000 exactly
    const int n0    = wave * 32;          // wave handles cols [n0, n0+32)

    const float* aRow = features + (size_t)(mBase + col) * IN_DIM;

    v8f c0 = {};
    v8f c1 = {};

    for (int k0 = 0; k0 < IN_DIM; k0 += 4) {
        const int kk = k0 + 2 * half;

        v2f afrag;
        { // two consecutive f32 along K of this lane's row
            const float2 av = *(const float2*)(aRow + kk);
            afrag.x = av.x; afrag.y = av.y;
        }

        const float* wp = W + (size_t)kk * OUT_DIM;   // rows kk, kk+1 of W
        v2f b0, b1;
        b0.x = wp[n0 + col];
        b0.y = wp[OUT_DIM + n0 + col];
        b1.x = wp[n0 + 16 + col];
        b1.y = wp[OUT_DIM + n0 + 16 + col];

        c0 = __builtin_amdgcn_wmma_f32_16x16x4_f32(false, afrag, false, b0,
                                                   (short)0, c0, false, false);
        c1 = __builtin_amdgcn_wmma_f32_16x16x4_f32(false, afrag, false, b1,
                                                   (short)0, c1, false, false);
    }

    // Epilogue: add bias, store
    #pragma unroll
    for (int v = 0; v < 8; ++v) {
        const int m = mBase + v + 8 * half;
        const int n = n0 + col;
        h[(size_t)m * OUT_DIM + n]      = c0[v] + bias[n];
        h[(size_t)m * OUT_DIM + n + 16] = c1[v] + bias[n + 16];
    }
}

// ---------------------------------------------------------------------------
// alpha1[n] = h[n] . a[0:128], alpha2[n] = h[n] . a[128:256]
// Wave-per-node: lane loads float4 (contiguous 512B per node), shfl reduce.
// ---------------------------------------------------------------------------
__global__ __launch_bounds__(256) void gat_alpha_kernel(
    const float* __restrict__ h,
    const float* __restrict__ a,          // [2*OUT_DIM]
    float* __restrict__ alpha1,
    float* __restrict__ alpha2)
{
    const int lane = threadIdx.x & 31;
    const int node = blockIdx.x * 8 + (threadIdx.x >> 5);
    if (node >= N_NODES) return;

    const float4 hx = ((const float4*)(h + (size_t)node * OUT_DIM))[lane];
    const float4 x1 = ((const float4*)(a))[lane];
    const float4 x2 = ((const float4*)(a + OUT_DIM))[lane];

    float s1 = hx.x * x1.x + hx.y * x1.y + hx.z * x1.z + hx.w * x1.w;
    float s2 = hx.x * x2.x + hx.y * x2.y + hx.z * x2.z + hx.w * x2.w;

    #pragma unroll
    for (int off = 16; off > 0; off >>= 1) {
        s1 += __shfl_xor(s1, off, 32);
        s2 += __shfl_xor(s2, off, 32);
    }
    if (lane == 0) {
        alpha1[node] = s1;
        alpha2[node] = s2;
    }
}

// ---------------------------------------------------------------------------
// Edge scatter: wave-per-edge. Lane l covers dims 4l..4l+3 -> 128 contiguous
// f32 atomics per edge (coalesced 512B burst into L2 atomic units).
// Virtual edges e >= N_EDGES are the self-loops (dst = src = e - N_EDGES).
// ---------------------------------------------------------------------------
__global__ __launch_bounds__(256) void gat_edge_kernel(
    const int*   __restrict__ edges,      // [N_EDGES, 2]
    const float* __restrict__ h,
    const float* __restrict__ alpha1,
    const float* __restrict__ alpha2,
    float* __restrict__ agg,              // == d_out [N_NODES, OUT_DIM]
    float* __restrict__ row_sum)          // [N_NODES]
{
    const int lane = threadIdx.x & 31;
    const long long e = (long long)blockIdx.x * 8 + (threadIdx.x >> 5);
    const long long TOT = (long long)N_EDGES + N_NODES;
    if (e >= TOT) return;

    int dst, src;
    if (e < N_EDGES) {
        dst = edges[2 * e];
        src = edges[2 * e + 1];
    } else {
        dst = src = (int)(e - N_EDGES);
    }

    const float logit = alpha1[dst] + alpha2[src];
    const float lr    = logit > 0.0f ? logit : SLOPE * logit;
    const float s     = __expf(lr);

    const float4 hx = ((const float4*)(h + (size_t)src * OUT_DIM))[lane];

    float* o = agg + (size_t)dst * OUT_DIM + lane * 4;
    atomicAdd(o + 0, s * hx.x);
    atomicAdd(o + 1, s * hx.y);
    atomicAdd(o + 2, s * hx.z);
    atomicAdd(o + 3, s * hx.w);

    if (lane == 0) atomicAdd(row_sum + dst, s);
}

// ---------------------------------------------------------------------------
// out[n,d] /= (row_sum[n] == 0 ? 1 : row_sum[n])
// ---------------------------------------------------------------------------
__global__ __launch_bounds__(256) void gat_finalize_kernel(
    float* __restrict__ out,
    const float* __restrict__ row_sum)
{
    const long long i = (long long)blockIdx.x * blockDim.x + threadIdx.x;
    const long long n = (long long)N_NODES * OUT_DIM;
    if (i >= n) return;
    const float rs = row_sum[i >> 7];     // OUT_DIM == 128
    out[i] = out[i] / (rs == 0.0f ? 1.0f : rs);
}

// ---------------------------------------------------------------------------
extern "C" void kernel_launch(void* const* d_in, const int* in_sizes, int n_in,
                              void* d_out, int out_size, void* d_ws, size_t ws_size,
                              hipStream_t stream) {
    const float* features = (const float*)d_in[0];
    const float* W        = (const float*)d_in[1];
    const float* bias     = (const float*)d_in[2];
    const float* a        = (const float*)d_in[3];
    const int*   edges    = (const int*)d_in[4];

    float* out = (float*)d_out;           // doubles as agg accumulator

    // Workspace layout: h | alpha1 | alpha2 | row_sum
    char* ws = (char*)d_ws;
    const size_t H_BYTES = (size_t)N_NODES * OUT_DIM * sizeof(float); // 25.6 MB
    float* h       = (float*)ws;
    float* alpha1  = (float*)(ws + H_BYTES);
    float* alpha2  = alpha1 + N_NODES;
    float* row_sum = alpha2 + N_NODES;

    // 1) zero agg (d_out) and row_sum
    {
        const long long n = (long long)N_NODES * OUT_DIM;
        gat_zero_kernel<<<2048, 256, 0, stream>>>(out, n);
        gat_zero_kernel<<<64, 256, 0, stream>>>(row_sum, (long long)N_NODES);
    }

    // 2) h = features @ W + b  (WMMA fp32)
    gat_gemm_kernel<<<N_NODES / 16, 128, 0, stream>>>(features, W, bias, h);

    // 3) per-node attention scalars
    gat_alpha_kernel<<<N_NODES / 8, 256, 0, stream>>>(h, a, alpha1, alpha2);

    // 4) edge scatter (1.6M edges + 50K self-loops, 8 waves/block)
    {
        const long long tot = (long long)N_EDGES + N_NODES;   // 1,650,000
        const int blocks = (int)((tot + 7) / 8);
        gat_edge_kernel<<<blocks, 256, 0, stream>>>(edges, h, alpha1, alpha2,
                                                    out, row_sum);
    }

    // 5) normalize
    {
        const long long n = (long long)N_NODES * OUT_DIM;
        const int blocks = (int)((n + 255) / 256);
        gat_finalize_kernel<<<blocks, 256, 0, stream>>>(out, row_sum);
    }
}